// MolecularThermodynamics_49598282334841
// MI455X (gfx1250) — compile-verified
//
#include <hip/hip_runtime.h>
#include <math.h>

// CDNA5 fp32 WMMA fragments: A/B 16x4 f32 = 2 VGPRs/lane, C/D 16x16 f32 = 8 VGPRs/lane.
typedef __attribute__((ext_vector_type(2))) float v2f;
typedef __attribute__((ext_vector_type(8))) float v8f;
typedef unsigned int uint32x4 __attribute__((ext_vector_type(4)));
typedef int int32x4 __attribute__((ext_vector_type(4)));
typedef int int32x8 __attribute__((ext_vector_type(8)));

#define LDP 68   // LDS row pitch in floats (64 data + 4 pad; 16B-aligned rows, no 64-bank conflicts)

__device__ __forceinline__ v8f wmma4(v2f a, v2f b, v8f c) {
  return __builtin_amdgcn_wmma_f32_16x16x4_f32(false, a, false, b, (short)0, c, false, false);
}

// Low 32 bits of a generic pointer to LDS = wave-relative LDS byte address.
__device__ __forceinline__ unsigned lds_addr(const void* p) {
  return (unsigned)(unsigned long long)(uintptr_t)p;
}

// TDM: DMA a row-major [rows x 64] fp32 tile from global into LDS, inserting
// 4 DWORDs of padding after every 64 DWORDs stored (-> stride-LDP layout).
__device__ __forceinline__ void tdm_load_rows64(unsigned lds_off, const float* gsrc, int rows) {
  unsigned long long ga = (unsigned long long)(uintptr_t)gsrc;
  uint32x4 g0;
  g0[0] = 1u;                                            // count=1, user descriptor
  g0[1] = lds_off;                                       // lds_addr (bytes)
  g0[2] = (unsigned)(ga & 0xFFFFFFFFu);                  // global_addr[31:0]
  g0[3] = (unsigned)((ga >> 32) & 0x01FFFFFFu) | (2u << 30);  // global_addr[56:32] | type=2
  int32x8 g1;
  // data_size=4B(2), pad_enable, pad_interval=64DW(5), pad_amount=4DW(3)
  g1[0] = (int)((2u << 16) | (1u << 20) | (5u << 22) | (3u << 25));
  g1[1] = (int)(64u << 16);                              // tensor_dim0 = 64
  g1[2] = (int)((unsigned)rows << 16);                   // tensor_dim1 = rows
  g1[3] = (int)(64u << 16);                              // tile_dim0 = 64
  g1[4] = rows;                                          // tile_dim1 = rows, tile_dim2 = 0
  g1[5] = 64;                                            // tensor_dim0_stride = 64
  g1[6] = 0;                                             // dim1 stride (unused, 2D)
  g1[7] = 0;
  int32x4 z4 = {0, 0, 0, 0};
#if __clang_major__ >= 23
  int32x8 z8 = {0, 0, 0, 0, 0, 0, 0, 0};
  __builtin_amdgcn_tensor_load_to_lds(g0, g1, z4, z4, z8, 0);
#else
  __builtin_amdgcn_tensor_load_to_lds(g0, g1, z4, z4, 0);
#endif
}

__device__ __forceinline__ float blockReduce128(float v, float* red) {
  const int t = threadIdx.x;
  __syncthreads();
  red[t] = v;
  __syncthreads();
  for (int s = 64; s > 0; s >>= 1) {
    if (t < s) red[t] += red[t + s];
    __syncthreads();
  }
  return red[0];
}

// ---------------------------------------------------------------------------
// Kernel A: embedding gather (x output), atomic energy/entropy linears,
// LJ vdW over i<j pairs, positional variance -> entropy. Block/molecule.
// ---------------------------------------------------------------------------
__global__ __launch_bounds__(128) void k_atoms(
    const int* __restrict__ atom_types, const float* __restrict__ positions,
    const float* __restrict__ emb, const float* __restrict__ ae_w,
    const float* __restrict__ ae_b, const float* __restrict__ ent_w,
    const float* __restrict__ ent_b, float* __restrict__ xout,
    float* __restrict__ epart, float* __restrict__ ent_out) {
  __shared__ int   ty[128];
  __shared__ float ps[128 * 3];
  __shared__ float red[128];
  __shared__ float sumx[64];
  const int b = blockIdx.x, t = threadIdx.x;

  ty[t] = atom_types[b * 128 + t];
  ps[t * 3 + 0] = positions[(b * 128 + t) * 3 + 0];
  ps[t * 3 + 1] = positions[(b * 128 + t) * 3 + 1];
  ps[t * 3 + 2] = positions[(b * 128 + t) * 3 + 2];
  __syncthreads();

  float colsum = 0.f;
  float* xb = xout + (size_t)b * 8192;
  for (int i = 0; i < 64; ++i) {
    int idx = t + 128 * i;              // coalesced; thread t always owns d = t&63
    int n = idx >> 6, d = idx & 63;
    float v = emb[ty[n] * 64 + d];
    xb[idx] = v;
    colsum += v;
  }
  __syncthreads();
  red[t] = colsum;
  __syncthreads();
  if (t < 64) sumx[t] = red[t] + red[t + 64];
  __syncthreads();

  float ae = 0.f, se = 0.f;
  if (t < 64) { ae = sumx[t] * ae_w[t]; se = sumx[t] * ent_w[t]; }
  float aeS = blockReduce128(ae, red);
  float seS = blockReduce128(se, red);

  float px = ps[t * 3], py = ps[t * 3 + 1], pz = ps[t * 3 + 2];
  float vdw = 0.f;
  for (int j = t + 1; j < 128; ++j) {
    float dx = px - ps[j * 3], dy = py - ps[j * 3 + 1], dz = pz - ps[j * 3 + 2];
    float d2 = dx * dx + dy * dy + dz * dz;
    if (d2 > 0.f) {
      float inv2 = 1.f / d2;
      float inv6 = inv2 * inv2 * inv2;
      vdw += 0.4f * (inv6 * inv6 - inv6);
    }
  }
  float vdwS = blockReduce128(vdw, red);
  float sx  = blockReduce128(px, red);
  float sy  = blockReduce128(py, red);
  float sz  = blockReduce128(pz, red);
  float sxx = blockReduce128(px * px, red);
  float syy = blockReduce128(py * py, red);
  float szz = blockReduce128(pz * pz, red);

  if (t == 0) {
    const float Nf = 128.f;
    float varx = (sxx - sx * sx / Nf) / (Nf - 1.f);
    float vary = (syy - sy * sy / Nf) / (Nf - 1.f);
    float varz = (szz - sz * sz / Nf) / (Nf - 1.f);
    epart[b]   = aeS + Nf * ae_b[0] + vdwS;
    ent_out[b] = seS + Nf * ent_b[0] + logf(1.f + varx + vary + varz);
  }
}

// ---------------------------------------------------------------------------
// Kernel B: bond bilinear energies. TDM DMAs X (with padding) and W_k into
// LDS; per k: T = X@Wk then E = T@X^T via fp32 WMMA with fused masked sum.
// Block = 256 threads (8 waves) per molecule. Wave w owns M-strip w*16..w*16+15
// in both GEMMs, so A-fragments are loaded once per wave per GEMM.
// ---------------------------------------------------------------------------
__global__ __launch_bounds__(256) void k_bond(
    const int* __restrict__ bonds, const float* __restrict__ xbuf,
    const float* __restrict__ bW, const float* __restrict__ bb,
    const float* __restrict__ epart, float* __restrict__ energy) {
  extern __shared__ float smem[];
  float* Xs  = smem;                 // 128 x LDP
  float* Ts  = Xs + 128 * LDP;       // 128 x LDP
  float* Wk  = Ts + 128 * LDP;       // 64 x LDP
  float* red = Wk + 64 * LDP;        // 256

  const int b = blockIdx.x, t = threadIdx.x;
  const int lane = t & 31, wave = t >> 5;
  const int hlf = lane >> 4, col = lane & 15;
  const float* xb = xbuf + (size_t)b * 8192;
  const int*   bo = bonds + (size_t)b * 16384;

  if (wave == 0) tdm_load_rows64(lds_addr(Xs), xb, 128);   // TENSORcnt++

  float local = 0.f;
  for (int k = 0; k < 4; ++k) {
    if (wave == 0) {
      tdm_load_rows64(lds_addr(Wk), bW + k * 4096, 64);
      __builtin_amdgcn_s_wait_tensorcnt(0);
    }
    __syncthreads();                 // TDM data visible to all waves

    // GEMM1: Ts = Xs(128x64) @ Wk(64x64). Wave w: mb = w*16, nb = q*16.
    {
      const int mb = wave * 16;
      v2f afr[16];
#pragma unroll
      for (int kk = 0; kk < 16; ++kk)
        afr[kk] = *(const v2f*)&Xs[(mb + col) * LDP + kk * 4 + 2 * hlf];
#pragma unroll
      for (int q = 0; q < 4; ++q) {
        const int nb = q * 16;
        v8f acc = {0.f, 0.f, 0.f, 0.f, 0.f, 0.f, 0.f, 0.f};
#pragma unroll
        for (int kk = 0; kk < 16; ++kk) {
          v2f bf;
          bf.x = Wk[(kk * 4 + 2 * hlf) * LDP + nb + col];
          bf.y = Wk[(kk * 4 + 2 * hlf + 1) * LDP + nb + col];
          acc = wmma4(afr[kk], bf, acc);
        }
#pragma unroll
        for (int r = 0; r < 8; ++r)
          Ts[(mb + r + 8 * hlf) * LDP + nb + col] = acc[r];
      }
    }
    __syncthreads();

    // GEMM2: E = Ts(128x64) @ Xs^T(64x128); masked epilogue only.
    {
      const float bbk = bb[k];
      const int mb = wave * 16;
      v2f afr[16];
#pragma unroll
      for (int kk = 0; kk < 16; ++kk)
        afr[kk] = *(const v2f*)&Ts[(mb + col) * LDP + kk * 4 + 2 * hlf];
#pragma unroll
      for (int q = 0; q < 8; ++q) {
        const int nb = q * 16;
        v8f acc = {0.f, 0.f, 0.f, 0.f, 0.f, 0.f, 0.f, 0.f};
#pragma unroll
        for (int kk = 0; kk < 16; ++kk) {
          // B[kk'][n] = X[nb+n][...]: contiguous pair, same pattern as A-frag.
          v2f bf = *(const v2f*)&Xs[(nb + col) * LDP + kk * 4 + 2 * hlf];
          acc = wmma4(afr[kk], bf, acc);
        }
        const int j = nb + col;
#pragma unroll
        for (int r = 0; r < 8; ++r) {
          int i = mb + r + 8 * hlf;
          if (j > i && bo[i * 128 + j] == k + 1) local += acc[r] + bbk;
        }
      }
    }
    __syncthreads();                 // GEMM2 reads done before next Ts/Wk writes
  }

  red[t] = local;
  __syncthreads();
  for (int s = 128; s > 0; s >>= 1) {
    if (t < s) red[t] += red[t + s];
    __syncthreads();
  }
  if (t == 0) energy[b] = epart[b] + red[0];
}

// ---------------------------------------------------------------------------
// Kernel C1: H = relu(mol @ W1 + b1), mol=[512,8192], W1=[8192,256].
// Double-buffered async (ASYNCcnt) global->LDS B128 copies overlap the next
// K-chunk's DMA with the current chunk's WMMAs. Block = 4 waves, 64x64 tile.
// ---------------------------------------------------------------------------
__global__ __launch_bounds__(128) void k_mlp1(
    const float* __restrict__ x, const float* __restrict__ sW1,
    const float* __restrict__ sb1, const float* __restrict__ dW1,
    const float* __restrict__ db1, float* __restrict__ h1s,
    float* __restrict__ h1d) {
  extern __shared__ float smem1[];           // 4 x (64*LDP) floats
  const int z = blockIdx.z;
  const float* W    = z ? dW1 : sW1;
  const float* bias = z ? db1 : sb1;
  float*       H    = z ? h1d : h1s;
  const int m0 = blockIdx.x * 64;
  const int n0 = blockIdx.y * 64;
  const int t = threadIdx.x;
  const int lane = t & 31, wave = t >> 5;
  const int hlf = lane >> 4, col = lane & 15;
  const unsigned CHUNK = 64u * LDP * 4u;     // bytes per LDS buffer
  const unsigned base = lds_addr(smem1);

  v8f acc[4];
#pragma unroll
  for (int q = 0; q < 4; ++q) acc[q] = (v8f){0.f, 0.f, 0.f, 0.f, 0.f, 0.f, 0.f, 0.f};

  auto issue_chunk = [&](int c, int buf) {
    const unsigned aOff = base + (unsigned)buf * 2u * CHUNK;
    const unsigned bOff = aOff + CHUNK;
#pragma unroll
    for (int it = 0; it < 8; ++it) {
      int e = (it * 128 + t) * 4;            // float index, 16B granules
      int r = e >> 6, cc = e & 63;
      unsigned lo = (unsigned)((r * LDP + cc) * 4);
      const float* gpA = x + (size_t)(m0 + r) * 8192 + c * 64 + cc;
      const float* gpB = W + (size_t)(c * 64 + r) * 256 + n0 + cc;
      asm volatile("global_load_async_to_lds_b128 %0, %1, off"
                   :: "v"(aOff + lo), "v"(gpA) : "memory");
      asm volatile("global_load_async_to_lds_b128 %0, %1, off"
                   :: "v"(bOff + lo), "v"(gpB) : "memory");
    }
  };

  issue_chunk(0, 0);
  asm volatile("s_wait_asynccnt 0x0" ::: "memory");
  __syncthreads();

  for (int c = 0; c < 128; ++c) {            // K = 8192 in chunks of 64
    const int cur = c & 1;
    if (c + 1 < 128) issue_chunk(c + 1, cur ^ 1);   // overlap DMA with compute
    const float* As = smem1 + (size_t)cur * 2 * 64 * LDP;
    const float* Bs = As + 64 * LDP;

    const int mb = wave * 16;
    v2f afr[16];
#pragma unroll
    for (int kk = 0; kk < 16; ++kk)
      afr[kk] = *(const v2f*)&As[(mb + col) * LDP + kk * 4 + 2 * hlf];
#pragma unroll
    for (int q = 0; q < 4; ++q) {
      const int nb = q * 16;
      v8f a = acc[q];
#pragma unroll
      for (int kk = 0; kk < 16; ++kk) {
        v2f bf;
        bf.x = Bs[(kk * 4 + 2 * hlf) * LDP + nb + col];
        bf.y = Bs[(kk * 4 + 2 * hlf + 1) * LDP + nb + col];
        a = wmma4(afr[kk], bf, a);
      }
      acc[q] = a;
    }
    asm volatile("s_wait_asynccnt 0x0" ::: "memory");
    __syncthreads();
  }

#pragma unroll
  for (int q = 0; q < 4; ++q) {
#pragma unroll
    for (int r = 0; r < 8; ++r) {
      int row = m0 + wave * 16 + r + 8 * hlf;
      int cg  = n0 + q * 16 + col;
      float v = acc[q][r] + bias[cg];
      H[(size_t)row * 256 + cg] = v > 0.f ? v : 0.f;
    }
  }
}

// ---------------------------------------------------------------------------
// Kernel C2: MLP layers 2+3 + sigmoid (34 MFLOP total -> VALU).
// ---------------------------------------------------------------------------
__global__ __launch_bounds__(128) void k_mlp_tail(
    const float* __restrict__ h1s, const float* __restrict__ h1d,
    const float* __restrict__ sW2, const float* __restrict__ sb2,
    const float* __restrict__ sW3, const float* __restrict__ sb3,
    const float* __restrict__ dW2, const float* __restrict__ db2,
    const float* __restrict__ dW3, const float* __restrict__ db3,
    float* __restrict__ outS, float* __restrict__ outD) {
  __shared__ float h[256];
  __shared__ float red[128];
  const int b = blockIdx.x, t = threadIdx.x, z = blockIdx.y;
  const float* h1 = (z ? h1d : h1s) + (size_t)b * 256;
  const float* W2 = z ? dW2 : sW2;
  const float* b2 = z ? db2 : sb2;
  const float* W3 = z ? dW3 : sW3;
  const float* b3 = z ? db3 : sb3;

  h[t] = h1[t];
  h[t + 128] = h1[t + 128];
  __syncthreads();

  float acc = b2[t];
  for (int kk = 0; kk < 256; ++kk) acc += h[kk] * W2[kk * 128 + t];
  acc = acc > 0.f ? acc : 0.f;

  red[t] = acc * W3[t];
  __syncthreads();
  for (int s = 64; s > 0; s >>= 1) {
    if (t < s) red[t] += red[t + s];
    __syncthreads();
  }
  if (t == 0) {
    float v = red[0] + b3[0];
    (z ? outD : outS)[b] = 1.f / (1.f + expf(-v));
  }
}

// ---------------------------------------------------------------------------
extern "C" void kernel_launch(void* const* d_in, const int* in_sizes, int n_in,
                              void* d_out, int out_size, void* d_ws, size_t ws_size,
                              hipStream_t stream) {
  (void)in_sizes; (void)n_in; (void)out_size; (void)ws_size;
  const int*   atom_types = (const int*)  d_in[0];
  const float* positions  = (const float*)d_in[1];
  const int*   bonds      = (const int*)  d_in[2];
  const float* emb   = (const float*)d_in[3];
  const float* ae_w  = (const float*)d_in[4];
  const float* ae_b  = (const float*)d_in[5];
  const float* ent_w = (const float*)d_in[6];
  const float* ent_b = (const float*)d_in[7];
  const float* bW    = (const float*)d_in[8];
  const float* bb    = (const float*)d_in[9];
  const float* sW1 = (const float*)d_in[10];
  const float* sb1 = (const float*)d_in[11];
  const float* sW2 = (const float*)d_in[12];
  const float* sb2 = (const float*)d_in[13];
  const float* sW3 = (const float*)d_in[14];
  const float* sb3 = (const float*)d_in[15];
  const float* dW1 = (const float*)d_in[16];
  const float* db1 = (const float*)d_in[17];
  const float* dW2 = (const float*)d_in[18];
  const float* db2 = (const float*)d_in[19];
  const float* dW3 = (const float*)d_in[20];
  const float* db3 = (const float*)d_in[21];

  float* out     = (float*)d_out;
  float* energy  = out;          // [512]
  float* entropy = out + 512;    // [512]
  float* stab    = out + 1024;   // [512]
  float* drug    = out + 1536;   // [512]
  float* xout    = out + 2048;   // [512*128*64]

  float* wsf   = (float*)d_ws;
  float* epart = wsf;                    // [512]
  float* h1s   = wsf + 512;              // [512*256]
  float* h1d   = h1s + 512 * 256;        // [512*256]

  k_atoms<<<512, 128, 0, stream>>>(atom_types, positions, emb, ae_w, ae_b,
                                   ent_w, ent_b, xout, epart, entropy);

  const size_t smemB = (size_t)(128 * LDP * 2 + 64 * LDP + 256) * sizeof(float);
  hipFuncSetAttribute((const void*)k_bond,
                      hipFuncAttributeMaxDynamicSharedMemorySize, (int)smemB);
  k_bond<<<512, 256, smemB, stream>>>(bonds, xout, bW, bb, epart, energy);

  const size_t smemC = (size_t)(4 * 64 * LDP) * sizeof(float);
  hipFuncSetAttribute((const void*)k_mlp1,
                      hipFuncAttributeMaxDynamicSharedMemorySize, (int)smemC);
  k_mlp1<<<dim3(8, 4, 2), 128, smemC, stream>>>(xout, sW1, sb1, dW1, db1, h1s, h1d);

  k_mlp_tail<<<dim3(512, 2), 128, 0, stream>>>(h1s, h1d, sW2, sb2, sW3, sb3,
                                               dW2, db2, dW3, db3, stab, drug);
}